// ExponentialMovingAverage_53730040873505
// MI455X (gfx1250) — compile-verified
//
#include <hip/hip_runtime.h>
#include <cstdint>
#include <cstddef>

// Problem constants (from reference): x[B,C,T] f32 -> out[B,T,C] f32
#define B_ 64
#define C_ 256
#define T_ 2048

#define CB 32                 // channels per block (== threads per block, 1 wave32)
#define TT 64                 // time-tile length
#define STRIDE (TT + 4)       // padded row stride in floats (rows stay 16B-aligned)
#define NT (T_ / TT)          // 32 tiles
#define NTHREADS CB

// Low 32 bits of a generic pointer to LDS == LDS byte offset (flat LDS aperture
// truncation per CDNA5 ISA).
__device__ __forceinline__ uint32_t lds_off(const void* p) {
    return (uint32_t)(uintptr_t)p;
}

// CDNA5 async global -> LDS copy, 16B per lane, tracked by ASYNCcnt.
__device__ __forceinline__ void async_load_b128(uint32_t lds_addr, const float* gaddr) {
    asm volatile("global_load_async_to_lds_b128 %0, %1, off"
                 :: "v"(lds_addr), "v"(gaddr)
                 : "memory");
}

__device__ __forceinline__ void wait_async0() {
    asm volatile("s_wait_asynccnt 0" ::: "memory");
}

__global__ __launch_bounds__(NTHREADS)
void ema_scan_kernel(const float* __restrict__ x,
                     const float* __restrict__ weights,
                     float* __restrict__ out) {
    // Double-buffered input tile: [2][CB rows][STRIDE floats] = 2*32*68*4 = 17408 B
    __shared__ __align__(16) float smem[2][CB * STRIDE];

    const int tid = threadIdx.x;                 // 0..31, one channel per lane
    const int b   = blockIdx.x >> 3;             // C_/CB == 8 channel blocks per batch
    const int c0  = (blockIdx.x & 7) * CB;
    const int c   = c0 + tid;

    float w = weights[c];
    w = fminf(fmaxf(w, 0.0f), 0.2f);
    const float om = 1.0f - w;

    const float* xbase = x + ((size_t)b * C_ + c0) * T_;   // this block's channel rows
    float* outp = out + (size_t)b * T_ * C_ + c;           // stride C_ between t's

    // Async-load mapping: 16 lanes cover one 64-float row (t4 = 0,4,...,60) as
    // b128s; 2 row-groups per pass (crow0 = tid>>4), 16 passes cover 32 rows.
    const int t4    = (tid & 15) * 4;  // float offset within row
    const int crow0 = tid >> 4;        // 0..1

    // ---- issue tile 0 into buffer 0 ----
    {
        const float* g = xbase + (size_t)crow0 * T_ + t4;
        uint32_t     l = lds_off(&smem[0][crow0 * STRIDE + t4]);
        #pragma unroll
        for (int it = 0; it < 16; ++it) {
            async_load_b128(l, g);
            g += 2 * (size_t)T_;       // 2 channel rows ahead
            l += 2 * STRIDE * 4;       // bytes
        }
    }
    wait_async0();                     // whole tile loaded by this wave's own ops
    __syncthreads();                   // single-wave WG: S_NOP in HW, compiler fence only

    float a = 0.0f;
    for (int k = 0; k < NT; ++k) {
        const int cur = k & 1;

        // ---- prefetch tile k+1 into the other buffer (consumed last iteration) ----
        if (k + 1 < NT) {
            const float* g = xbase + (size_t)crow0 * T_ + (k + 1) * TT + t4;
            uint32_t     l = lds_off(&smem[cur ^ 1][crow0 * STRIDE + t4]);
            #pragma unroll
            for (int it = 0; it < 16; ++it) {
                async_load_b128(l, g);
                g += 2 * (size_t)T_;
                l += 2 * STRIDE * 4;
            }
        }

        // ---- sequential EMA scan over this tile from LDS (ds_load_b128 reads) ----
        const float4* buf4 =
            reinterpret_cast<const float4*>(&smem[cur][tid * STRIDE]);
        const int tbase = k * TT;
        int j4 = 0;
        if (k == 0) {
            float4 v = buf4[0];
            a = v.x;                     outp[0]              = a;  // t=0: seed, no EMA
            a = fmaf(w, v.y, om * a);    outp[(size_t)1 * C_] = a;
            a = fmaf(w, v.z, om * a);    outp[(size_t)2 * C_] = a;
            a = fmaf(w, v.w, om * a);    outp[(size_t)3 * C_] = a;
            j4 = 1;
        }
        for (; j4 < TT / 4; ++j4) {
            float4 v = buf4[j4];
            const size_t t = (size_t)(tbase + j4 * 4);
            a = fmaf(w, v.x, om * a);    outp[(t + 0) * C_] = a;
            a = fmaf(w, v.y, om * a);    outp[(t + 1) * C_] = a;
            a = fmaf(w, v.z, om * a);    outp[(t + 2) * C_] = a;
            a = fmaf(w, v.w, om * a);    outp[(t + 3) * C_] = a;
        }

        // Wait for this wave's prefetch before scanning it next iteration.
        // All ds_load results above were consumed (s_wait_dscnt inserted by the
        // compiler before the FMAs), so re-targeting the old buffer is safe.
        wait_async0();
        __syncthreads();
    }
}

extern "C" void kernel_launch(void* const* d_in, const int* in_sizes, int n_in,
                              void* d_out, int out_size, void* d_ws, size_t ws_size,
                              hipStream_t stream) {
    (void)in_sizes; (void)n_in; (void)out_size; (void)d_ws; (void)ws_size;
    const float* x   = (const float*)d_in[0];
    // d_in[1] = initial_state: accepted but unused (reference seeds with x[:, :, 0])
    const float* wts = (const float*)d_in[2];
    float* out = (float*)d_out;

    dim3 grid(B_ * (C_ / CB));   // 512 single-wave workgroups -> max WGP spread
    dim3 block(NTHREADS);        // 32 threads = 1 wave32
    ema_scan_kernel<<<grid, block, 0, stream>>>(x, wts, out);
}